// original_gcnalign_with_inputs_77163382440897
// MI455X (gfx1250) — compile-verified
//
#include <hip/hip_runtime.h>

#define N_NODES 100000
#define N_EDGES 1600000
#define D 128

typedef __attribute__((ext_vector_type(2))) float v2f;
typedef __attribute__((ext_vector_type(8))) float v8f;

// ---------------------------------------------------------------------------
// h = x @ w   (fp32, exact, via V_WMMA_F32_16X16X4_F32)
// One block = 8 waves = one 16-row tile; wave j owns columns [16j, 16j+16).
// K=128 consumed in 32 steps of 4.
// ---------------------------------------------------------------------------
__global__ __launch_bounds__(256) void gemm_xw(const float* __restrict__ x,
                                               const float* __restrict__ w,
                                               float* __restrict__ h) {
    __shared__ float wlds[D * D];   // 64 KB: full weight matrix staged in LDS
    {
        const float4* wsrc = reinterpret_cast<const float4*>(w);
        float4*       wdst = reinterpret_cast<float4*>(wlds);
        for (int i = threadIdx.x; i < (D * D) / 4; i += 256)
            wdst[i] = wsrc[i];
    }
    __syncthreads();

    const int lane    = threadIdx.x & 31;
    const int wave    = threadIdx.x >> 5;   // 0..7 -> column tile
    const int lo      = lane & 15;
    const int hi      = lane >> 4;          // 0 or 1 (half-wave)
    const int rowBase = blockIdx.x * 16;    // 100000/16 = 6250 blocks exactly
    const int colBase = wave * 16;

    // A (16x4 f32) layout: lanes 0-15 M=lane {VGPR0:K=0, VGPR1:K=1},
    //                      lanes 16-31 M=lane-16 {K=2, K=3}
    const float* arow = x + (size_t)(rowBase + lo) * D;

    v8f c = {0.f, 0.f, 0.f, 0.f, 0.f, 0.f, 0.f, 0.f};
    for (int k = 0; k < D; k += 4) {
        const int ka = k + hi * 2;
        const float2 a2 = *reinterpret_cast<const float2*>(arow + ka);
        v2f a; a.x = a2.x; a.y = a2.y;
        // B (4x16 f32), mirror layout: lanes 0-15 N=lane {K=0,K=1},
        //                              lanes 16-31 N=lane-16 {K=2,K=3}
        v2f b;
        b.x = wlds[(ka + 0) * D + colBase + lo];
        b.y = wlds[(ka + 1) * D + colBase + lo];
        c = __builtin_amdgcn_wmma_f32_16x16x4_f32(
                /*neg_a=*/false, a, /*neg_b=*/false, b,
                /*c_mod=*/(short)0, c, /*reuse_a=*/false, /*reuse_b=*/false);
    }

    // C/D 16x16 f32 layout: VGPR v -> M = v + 8*hi, N = lo
    float* hout = h + (size_t)(rowBase + hi * 8) * D + colBase + lo;
#pragma unroll
    for (int v = 0; v < 8; ++v)
        hout[(size_t)v * D] = c[v];
}

// ---------------------------------------------------------------------------
// dst[r] += val * src[c] per edge — one wave32 per edge, float4 per lane,
// native fp32 HW atomics (global_atomic_add_f32).
// ---------------------------------------------------------------------------
__global__ __launch_bounds__(256) void spmm_edges(const int* __restrict__ rows,
                                                  const int* __restrict__ cols,
                                                  const float* __restrict__ vals,
                                                  const float* __restrict__ src,
                                                  float* __restrict__ dst) {
    const int e = (int)((blockIdx.x * 256u + threadIdx.x) >> 5);
    if (e >= N_EDGES) return;
    const int   lane = threadIdx.x & 31;
    const int   r    = rows[e];
    const int   c    = cols[e];
    const float v    = vals[e];

    const float4 hv =
        *reinterpret_cast<const float4*>(src + (size_t)c * D + lane * 4);
    float* o = dst + (size_t)r * D + lane * 4;
    unsafeAtomicAdd(o + 0, v * hv.x);
    unsafeAtomicAdd(o + 1, v * hv.y);
    unsafeAtomicAdd(o + 2, v * hv.z);
    unsafeAtomicAdd(o + 3, v * hv.w);
}

__global__ __launch_bounds__(256) void zero_f4(float4* __restrict__ p, int n4) {
    const int i = blockIdx.x * 256 + threadIdx.x;
    if (i < n4) p[i] = make_float4(0.f, 0.f, 0.f, 0.f);
}

extern "C" void kernel_launch(void* const* d_in, const int* in_sizes, int n_in,
                              void* d_out, int out_size, void* d_ws, size_t ws_size,
                              hipStream_t stream) {
    const float* x  = (const float*)d_in[0];   // [N_NODES, D]
    const float* w  = (const float*)d_in[1];   // [D, D]
    const int*   er = (const int*)d_in[2];     // [N_EDGES]
    const int*   ec = (const int*)d_in[3];     // [N_EDGES]
    const float* ev = (const float*)d_in[4];   // [N_EDGES]
    float*       out = (float*)d_out;          // [N_NODES, D]

    float* h = (float*)d_ws;                   // 51.2 MB
    float* y = h + (size_t)N_NODES * D;        // 51.2 MB

    const int n4 = N_NODES * D / 4;
    zero_f4<<<(n4 + 255) / 256, 256, 0, stream>>>((float4*)y, n4);
    zero_f4<<<(n4 + 255) / 256, 256, 0, stream>>>((float4*)out, n4);

    gemm_xw<<<N_NODES / 16, 256, 0, stream>>>(x, w, h);

    const int spmm_blocks = (N_EDGES * 32 + 255) / 256;  // 1 wave per edge
    spmm_edges<<<spmm_blocks, 256, 0, stream>>>(er, ec, ev, h, y);
    spmm_edges<<<spmm_blocks, 256, 0, stream>>>(er, ec, ev, y, out);
}